// SNNEncode_63479616635534
// MI455X (gfx1250) — compile-verified
//
#include <hip/hip_runtime.h>

typedef __attribute__((ext_vector_type(2))) float v2f;
typedef __attribute__((ext_vector_type(4))) float v4f;
typedef __attribute__((ext_vector_type(8))) float v8f;

// Problem dimensions (fixed by the reference setup_inputs()).
#define T_STEPS 2000
#define B_DIM   128
#define IN_DIM  16
#define H_DIM   256
#define BH      (B_DIM * H_DIM)      // 32768 independent LIF states
#define V_TH    1.0f

// ---------------------------------------------------------------------------
// Phase 1: xw[t][b][h] = sum_i x[b][t][i] * W[h][i]  via V_WMMA_F32_16X16X4_F32
// One wave computes one 16(b) x 16(h) tile at a fixed t, chaining 4 k=4 WMMAs
// to cover K = IN = 16.  Tiles: T * (B/16) * (H/16) = 2000*8*16 = 256000 waves.
// ---------------------------------------------------------------------------
__global__ __launch_bounds__(256) void snn_proj_wmma(
    const float* __restrict__ x,    // [B, T, IN]
    const float* __restrict__ W,    // [H, IN]
    float* __restrict__ xw)         // [T, B, H]
{
    const int lane = threadIdx.x & 31;
    const int tile = blockIdx.x * 8 + (threadIdx.x >> 5);
    // tile = ((t * 8) + b_tile) * 16 + h_tile
    const int ht = tile & 15;
    const int bt = (tile >> 4) & 7;
    const int t  = tile >> 7;
    if (t >= T_STEPS) return;       // wave-uniform; EXEC stays all-1s for WMMA

    const int rc = lane & 15;            // A: row (b), B: col (h)
    const int kb = (lane >> 4) << 1;     // K sub-pair: lanes 0-15 -> k{0,1}, 16-31 -> k{2,3}
    const int b0 = bt * 16;
    const int h0 = ht * 16;

    // A row for this lane: x[b0+rc][t][*]; B col: W[h0+rc][*]
    const float* xrow = x + ((size_t)(b0 + rc) * T_STEPS + (size_t)t) * IN_DIM;
    const float* wrow = W + (size_t)(h0 + rc) * IN_DIM;

    v8f acc = {};
#pragma unroll
    for (int c = 0; c < 4; ++c) {        // K chunks of 4
        v2f a = *(const v2f*)(xrow + c * 4 + kb);
        v2f b = *(const v2f*)(wrow + c * 4 + kb);
        // (neg_a, A, neg_b, B, c_mod, C, reuse_a, reuse_b)
        acc = __builtin_amdgcn_wmma_f32_16x16x4_f32(
            false, a, false, b, (short)0, acc, false, false);
    }

    // D layout: VGPR r -> M = r + (lane/16)*8, N = lane%16
    float* outp = xw + (size_t)t * BH
                     + (size_t)(b0 + (lane >> 4) * 8) * H_DIM
                     + (size_t)(h0 + rc);
#pragma unroll
    for (int r = 0; r < 8; ++r)
        outp[(size_t)r * H_DIM] = acc[r];
}

// ---------------------------------------------------------------------------
// Phase 2: LIF scan. One thread per (b,h) state: 32768 threads = 1024 waves.
// Per timestep: 1 streaming load (xw), ~7 VALU, 1 streaming store (spike).
// Prefetch a few timesteps ahead to hide HBM latency on the serial chain.
// ---------------------------------------------------------------------------
__global__ __launch_bounds__(256) void snn_scan(
    const float* __restrict__ xw,        // [T, B, H]
    const float* __restrict__ tau_syn,   // [1, H]
    const float* __restrict__ tau_mem,   // [1, H]
    float* __restrict__ out)             // spikes [T,B,H] ++ v[B,H] ++ i[B,H]
{
    const int id = blockIdx.x * blockDim.x + threadIdx.x;   // [0, BH)
    const int h  = id & (H_DIM - 1);
    const float ts = fminf(fmaxf(tau_syn[h], 0.0f), 1.0f);
    const float tm = fminf(fmaxf(tau_mem[h], 0.0f), 1.0f);

    float v = 0.0f, cur = 0.0f;
    const float* xp = xw + id;
    float* sp = out + id;

    for (int t = 0; t < T_STEPS; ++t) {
        const float inp = xp[(size_t)t * BH];
        const int tp = (t + 8 < T_STEPS) ? t + 8 : (T_STEPS - 1);
        __builtin_prefetch(xp + (size_t)tp * BH, 0, 1);     // global_prefetch_b8

        const float v_dec = fmaf(tm, cur - v, v);           // v + tm*(i - v)
        const float i_dec = cur - ts * cur;                 // i*(1 - ts)
        const bool  fire  = (v_dec - V_TH) > 0.0f;          // heaviside
        sp[(size_t)t * BH] = fire ? 1.0f : 0.0f;
        v   = fire ? 0.0f : v_dec;                          // (1-z)*v_dec
        cur = i_dec + inp;
    }
    out[(size_t)T_STEPS * BH + id]      = v;                // v_f
    out[(size_t)T_STEPS * BH + BH + id] = cur;              // i_f
}

// ---------------------------------------------------------------------------
// Fallback: fused projection + scan (used only if d_ws can't hold xw).
// Each thread keeps its W row in registers; x[b][t][*] (64B) is broadcast-read
// by the 256 h-threads of the same b and stays hot in the WGP cache.
// ---------------------------------------------------------------------------
__global__ __launch_bounds__(256) void snn_fused(
    const float* __restrict__ x, const float* __restrict__ W,
    const float* __restrict__ tau_syn, const float* __restrict__ tau_mem,
    float* __restrict__ out)
{
    const int id = blockIdx.x * blockDim.x + threadIdx.x;
    const int h  = id & (H_DIM - 1);
    const int b  = id >> 8;
    const float ts = fminf(fmaxf(tau_syn[h], 0.0f), 1.0f);
    const float tm = fminf(fmaxf(tau_mem[h], 0.0f), 1.0f);

    float w[IN_DIM];
#pragma unroll
    for (int i = 0; i < IN_DIM; ++i) w[i] = W[h * IN_DIM + i];

    float v = 0.0f, cur = 0.0f;
    const float* xb = x + (size_t)b * T_STEPS * IN_DIM;
    float* sp = out + id;

    for (int t = 0; t < T_STEPS; ++t) {
        const float* xt = xb + (size_t)t * IN_DIM;
        float inp = 0.0f;
#pragma unroll
        for (int c = 0; c < 4; ++c) {
            const v4f xv = *(const v4f*)(xt + c * 4);
            inp = fmaf(xv.x, w[c * 4 + 0],
                  fmaf(xv.y, w[c * 4 + 1],
                  fmaf(xv.z, w[c * 4 + 2],
                  fmaf(xv.w, w[c * 4 + 3], inp))));
        }
        const float v_dec = fmaf(tm, cur - v, v);
        const float i_dec = cur - ts * cur;
        const bool  fire  = (v_dec - V_TH) > 0.0f;
        sp[(size_t)t * BH] = fire ? 1.0f : 0.0f;
        v   = fire ? 0.0f : v_dec;
        cur = i_dec + inp;
    }
    out[(size_t)T_STEPS * BH + id]      = v;
    out[(size_t)T_STEPS * BH + BH + id] = cur;
}

extern "C" void kernel_launch(void* const* d_in, const int* in_sizes, int n_in,
                              void* d_out, int out_size, void* d_ws, size_t ws_size,
                              hipStream_t stream) {
    const float* x       = (const float*)d_in[0];  // [128,2000,16]
    const float* W       = (const float*)d_in[1];  // [256,16]
    const float* tau_syn = (const float*)d_in[2];  // [1,256]
    const float* tau_mem = (const float*)d_in[3];  // [1,256]
    float* out = (float*)d_out;                    // spikes ++ v_f ++ i_f

    const size_t xw_bytes = (size_t)T_STEPS * BH * sizeof(float);   // 262 MB
    if (ws_size >= xw_bytes) {
        float* xw = (float*)d_ws;
        // 256000 tiles, 8 waves (one tile each) per 256-thread block
        snn_proj_wmma<<<(T_STEPS * 8 * 16) / 8, 256, 0, stream>>>(x, W, xw);
        snn_scan<<<BH / 256, 256, 0, stream>>>(xw, tau_syn, tau_mem, out);
    } else {
        snn_fused<<<BH / 256, 256, 0, stream>>>(x, W, tau_syn, tau_mem, out);
    }
}